// GraphSAGELayer_7894149890554
// MI455X (gfx1250) — compile-verified
//
#include <hip/hip_runtime.h>
#include <hip/hip_bf16.h>

#define NODES 50000
#define EDGES 800000
#define DIM 64
#define ROW_TILES (NODES / 16)   // 3125, exact

typedef __attribute__((ext_vector_type(2))) float v2f;
typedef __attribute__((ext_vector_type(8))) float v8f;

// ---------------------------------------------------------------------------
// Kernel 1: zero the workspace (agg[N*64] + deg[N])
// ---------------------------------------------------------------------------
__global__ void sage_zero_ws(float* __restrict__ ws, int n) {
    int i = blockIdx.x * blockDim.x + threadIdx.x;
    int stride = gridDim.x * blockDim.x;
    for (; i < n; i += stride) ws[i] = 0.0f;
}

// ---------------------------------------------------------------------------
// Kernel 2: edge scatter: agg[src] += x[dst]; deg[src] += 1
// 16 lanes per edge, each lane atomically adds 4 consecutive features.
// ---------------------------------------------------------------------------
__global__ void sage_scatter(const int* __restrict__ edge_index,
                             const float* __restrict__ x,
                             float* __restrict__ agg,
                             float* __restrict__ deg) {
    int t = blockIdx.x * blockDim.x + threadIdx.x;
    int edge = t >> 4;
    int sub = t & 15;
    if (edge >= EDGES) return;
    int src = edge_index[edge];          // edge_index[0][e]
    int dst = edge_index[EDGES + edge];  // edge_index[1][e]
    const float4 v = *(const float4*)(x + (size_t)dst * DIM + sub * 4);
    float* a = agg + (size_t)src * DIM + sub * 4;
    atomicAdd(a + 0, v.x);
    atomicAdd(a + 1, v.y);
    atomicAdd(a + 2, v.z);
    atomicAdd(a + 3, v.w);
    if (sub == 0) atomicAdd(deg + src, 1.0f);
}

// ---------------------------------------------------------------------------
// Kernel 3: fused dual-GEMM + bias + ReLU via V_WMMA_F32_16X16X4_F32.
// out = relu(x @ Ws^T + (agg/deg) @ Wn^T + b_self + b_neigh)
// One wave32 per 16-node tile; 4 column tiles of 16 (DIM=64).
// ---------------------------------------------------------------------------
__global__ __launch_bounds__(256) void sage_gemm_wmma(
    const float* __restrict__ x, const float* __restrict__ agg,
    const float* __restrict__ deg,
    const float* __restrict__ Ws, const float* __restrict__ Wn,
    const float* __restrict__ bs, const float* __restrict__ bn,
    float* __restrict__ out) {
    const int lane = threadIdx.x & 31;
    const int wave = threadIdx.x >> 5;                 // 0..7
    const int tile = blockIdx.x * 8 + wave;            // 16-row tile
    if (tile >= ROW_TILES) return;                     // wave-uniform: EXEC stays full
    const int row0 = tile * 16;
    const int hi = lane >> 4;                          // lane half: 0 or 1
    const int l  = lane & 15;

    const int arow = row0 + l;                         // row this lane feeds into A
    const float d = deg[arow];
    const float inv = 1.0f / fmaxf(d, 1.0f);

    // Preload A fragments (x row and normalized agg row) for all 16 k-chunks.
    // A layout (16x4 f32): VGPR0 = K=k0+2*hi, VGPR1 = K=k0+2*hi+1 -> float2.
    v2f ax[16], an[16];
#pragma unroll
    for (int k = 0; k < 16; ++k) {
        const int kk = k * 4 + 2 * hi;
        ax[k] = *(const v2f*)(x + (size_t)arow * DIM + kk);
        v2f t = *(const v2f*)(agg + (size_t)arow * DIM + kk);
        an[k].x = t.x * inv;
        an[k].y = t.y * inv;
    }

#pragma unroll
    for (int nt = 0; nt < 4; ++nt) {
        const int n0 = nt * 16;
        // Bias depends only on column N = l; broadcast into all 8 C rows.
        const float bias = bs[n0 + l] + bn[n0 + l];
        v8f c;
#pragma unroll
        for (int j = 0; j < 8; ++j) c[j] = bias;

#pragma unroll
        for (int k = 0; k < 16; ++k) {
            const int kk = k * 4 + 2 * hi;
            // B layout (4x16): B[k][n] = W[n][k]; lane (hi,l) loads
            // W[(n0+l)*DIM + kk .. kk+1] as float2 (contiguous).
            v2f bws = *(const v2f*)(Ws + (size_t)(n0 + l) * DIM + kk);
            v2f bwn = *(const v2f*)(Wn + (size_t)(n0 + l) * DIM + kk);
            c = __builtin_amdgcn_wmma_f32_16x16x4_f32(
                false, ax[k], false, bws, (short)0, c, false, false);
            c = __builtin_amdgcn_wmma_f32_16x16x4_f32(
                false, an[k], false, bwn, (short)0, c, false, false);
        }

        // D layout: c[j] -> out[row0 + j + 8*hi][n0 + l]; fuse ReLU.
#pragma unroll
        for (int j = 0; j < 8; ++j) {
            const int m = j + 8 * hi;
            out[(size_t)(row0 + m) * DIM + n0 + l] = fmaxf(c[j], 0.0f);
        }
    }
}

// ---------------------------------------------------------------------------
// Launcher
// ---------------------------------------------------------------------------
extern "C" void kernel_launch(void* const* d_in, const int* in_sizes, int n_in,
                              void* d_out, int out_size, void* d_ws, size_t ws_size,
                              hipStream_t stream) {
    const float* x  = (const float*)d_in[0];
    const int* ei   = (const int*)d_in[1];   // [2, E] flat
    const float* Ws = (const float*)d_in[2];
    const float* bs = (const float*)d_in[3];
    const float* Wn = (const float*)d_in[4];
    const float* bn = (const float*)d_in[5];
    float* out = (float*)d_out;

    float* agg = (float*)d_ws;               // N*64 floats
    float* deg = agg + (size_t)NODES * DIM;  // N floats

    // 1) zero agg + deg
    const int nz = NODES * DIM + NODES;
    sage_zero_ws<<<1024, 256, 0, stream>>>((float*)d_ws, nz);

    // 2) scatter-add over edges (16 lanes per edge)
    const int sthreads = EDGES * 16;
    sage_scatter<<<(sthreads + 255) / 256, 256, 0, stream>>>(ei, x, agg, deg);

    // 3) fused WMMA GEMM + bias + ReLU (8 waves = 8 row-tiles per block)
    const int blocks = (ROW_TILES + 7) / 8;
    sage_gemm_wmma<<<blocks, 256, 0, stream>>>(x, agg, deg, Ws, Wn, bs, bn, out);
}